// Net_SLSTM_87402584474195
// MI455X (gfx1250) — compile-verified
//
#include <hip/hip_runtime.h>
#include <math.h>

#define T_STEPS 256
#define LDIM    1024
#define CIN     14
#define CMID    32
#define HD      128
#define G4      512      // 4*HD gates
#define TL      16       // L-rows per block
#define NCLS    7

typedef __attribute__((ext_vector_type(2))) float v2f;
typedef __attribute__((ext_vector_type(8))) float v8f;

#define HAVE_WMMA_F32 __has_builtin(__builtin_amdgcn_wmma_f32_16x16x4_f32)

__device__ __forceinline__ float sigmoidf_(float v) { return 1.0f / (1.0f + expf(-v)); }

// gates[TL][512] = At[TL][IND1] @ W1[512][IND1]^T + MemS[TL][HD] @ W2[512][HD]^T
// At/MemS/Gates in LDS, W1/W2 streamed from global (L2-resident).
// k-outer / 4 N-tiles inner: one shared A-fragment per k-step feeding 4
// independent WMMA accumulator chains (fills the matrix pipe despite low
// wave occupancy of the inherently batch-limited recurrence).
template <int IND1>
__device__ __forceinline__ void gemm_gates(const float* At, const float* MemS,
                                           const float* __restrict__ W1,
                                           const float* __restrict__ W2,
                                           float* Gates)
{
  const int tid = threadIdx.x;
#if HAVE_WMMA_F32
  const int wave = tid >> 5;
  const int lane = tid & 31;
  const int m16  = lane & 15;      // A row / B-N col within tile
  const int kh   = lane >> 4;      // K half select
  const int ncol = wave * 64 + m16; // this wave owns gate cols [wave*64, wave*64+64)

  v8f acc0 = {}, acc1 = {}, acc2 = {}, acc3 = {};

  for (int kb = 0; kb < IND1; kb += 4) {
    const int k0 = kb + 2 * kh;
    const v2f a  = *(const v2f*)(At + m16 * IND1 + k0);
    const v2f b0 = *(const v2f*)(W1 + (size_t)(ncol +  0) * IND1 + k0);
    const v2f b1 = *(const v2f*)(W1 + (size_t)(ncol + 16) * IND1 + k0);
    const v2f b2 = *(const v2f*)(W1 + (size_t)(ncol + 32) * IND1 + k0);
    const v2f b3 = *(const v2f*)(W1 + (size_t)(ncol + 48) * IND1 + k0);
    acc0 = __builtin_amdgcn_wmma_f32_16x16x4_f32(false, a, false, b0, (short)0, acc0, false, false);
    acc1 = __builtin_amdgcn_wmma_f32_16x16x4_f32(false, a, false, b1, (short)0, acc1, false, false);
    acc2 = __builtin_amdgcn_wmma_f32_16x16x4_f32(false, a, false, b2, (short)0, acc2, false, false);
    acc3 = __builtin_amdgcn_wmma_f32_16x16x4_f32(false, a, false, b3, (short)0, acc3, false, false);
  }
  for (int kb = 0; kb < HD; kb += 4) {
    const int k0 = kb + 2 * kh;
    const v2f a  = *(const v2f*)(MemS + m16 * HD + k0);
    const v2f b0 = *(const v2f*)(W2 + (size_t)(ncol +  0) * HD + k0);
    const v2f b1 = *(const v2f*)(W2 + (size_t)(ncol + 16) * HD + k0);
    const v2f b2 = *(const v2f*)(W2 + (size_t)(ncol + 32) * HD + k0);
    const v2f b3 = *(const v2f*)(W2 + (size_t)(ncol + 48) * HD + k0);
    acc0 = __builtin_amdgcn_wmma_f32_16x16x4_f32(false, a, false, b0, (short)0, acc0, false, false);
    acc1 = __builtin_amdgcn_wmma_f32_16x16x4_f32(false, a, false, b1, (short)0, acc1, false, false);
    acc2 = __builtin_amdgcn_wmma_f32_16x16x4_f32(false, a, false, b2, (short)0, acc2, false, false);
    acc3 = __builtin_amdgcn_wmma_f32_16x16x4_f32(false, a, false, b3, (short)0, acc3, false, false);
  }
#pragma unroll
  for (int r = 0; r < 8; ++r) {            // D layout: reg r -> row r+8*kh, col = n
    const int row = (r + 8 * kh) * G4;
    Gates[row + ncol]      = acc0[r];
    Gates[row + ncol + 16] = acc1[r];
    Gates[row + ncol + 32] = acc2[r];
    Gates[row + ncol + 48] = acc3[r];
  }
#else
  for (int e = tid; e < TL * G4; e += 256) {
    const int r = e >> 9;              // /512
    const int n = e & (G4 - 1);
    float s = 0.f;
    for (int k = 0; k < IND1; ++k) s += At[r * IND1 + k] * W1[(size_t)n * IND1 + k];
    for (int k = 0; k < HD; ++k)   s += MemS[r * HD + k] * W2[(size_t)n * HD + k];
    Gates[e] = s;
  }
#endif
}

__global__ void zero_counts_kernel(unsigned int* counts) {
  if (threadIdx.x < HD) counts[threadIdx.x] = 0u;
}

// ---- Layer 1: fused conv1d + leaky-spike + SLSTM scan; emits packed spikes + per-channel counts
__global__ __launch_bounds__(256) void slstm1_kernel(
    const float* __restrict__ x,       // [T][L][14]
    const float* __restrict__ conv_w,  // [32][14][3]
    const float* __restrict__ conv_b,  // [32]
    const float* __restrict__ w_ih1,   // [512][32]
    const float* __restrict__ w_hh1,   // [512][128]
    const float* __restrict__ b_ih1,   // [512]
    const float* __restrict__ b_hh1,   // [512]
    const float* __restrict__ thr1p,   // [1]
    unsigned char* __restrict__ spk1,  // [T][L][128] (0/1 bytes)
    unsigned int* __restrict__ counts) // [128]
{
  __shared__ float At[TL * CMID];
  __shared__ float MemS[TL * HD];
  __shared__ float SynS[TL * HD];
  __shared__ float Gates[TL * G4];

  const int tid = threadIdx.x;
  const int l0  = blockIdx.x * TL;
  const float thr = thr1p[0];

  for (int e = tid; e < TL * HD; e += 256) { MemS[e] = 0.f; SynS[e] = 0.f; }

  // elementwise ownership: (h, rows rg*8..rg*8+7)
  const int h  = tid & (HD - 1);
  const int rg = tid >> 7;
  const float bi = b_ih1[h]            + b_hh1[h];
  const float bf = b_ih1[h + HD]       + b_hh1[h + HD];
  const float bg = b_ih1[h + 2 * HD]   + b_hh1[h + 2 * HD];
  const float bo = b_ih1[h + 3 * HD]   + b_hh1[h + 3 * HD];
  unsigned int cnt = 0;
  __syncthreads();

  for (int t = 0; t < T_STEPS; ++t) {
    // phase 1: conv(k=3,pad=1 over L) + heaviside -> At[TL][32]
    for (int j = 0; j < 2; ++j) {
      const int e = tid + j * 256;       // 0..511
      const int r = e >> 5;
      const int c = e & 31;
      const int l = l0 + r;
      float v = conv_b[c];
      for (int kk = 0; kk < 3; ++kk) {
        const int ll = l + kk - 1;
        if (ll >= 0 && ll < LDIM) {
          const float* xp = x + ((size_t)t * LDIM + ll) * CIN;
          const float* wp = conv_w + c * (CIN * 3) + kk;
          for (int ci = 0; ci < CIN; ++ci) v += xp[ci] * wp[ci * 3];
        }
      }
      At[r * CMID + c] = (v - 1.0f > 0.f) ? 1.0f : 0.0f;
    }
    __syncthreads();

    gemm_gates<CMID>(At, MemS, w_ih1, w_hh1, Gates);
    __syncthreads();

    // phase 3: LSTM state update + spike
    for (int jr = 0; jr < 8; ++jr) {
      const int r = rg * 8 + jr;
      const float gi = Gates[r * G4 + h]          + bi;
      const float gf = Gates[r * G4 + h + HD]     + bf;
      const float gg = Gates[r * G4 + h + 2 * HD] + bg;
      const float go = Gates[r * G4 + h + 3 * HD] + bo;
      const float syn  = SynS[r * HD + h];
      const float memv = MemS[r * HD + h];
      const float sy = sigmoidf_(gf) * syn + sigmoidf_(gi) * tanhf(gg);
      const float mn = sigmoidf_(go) * tanhf(sy) - ((memv - thr > 0.f) ? thr : 0.f);
      SynS[r * HD + h] = sy;
      MemS[r * HD + h] = mn;
      const unsigned char s = (mn - thr > 0.f) ? 1 : 0;
      spk1[((size_t)t * LDIM + (l0 + r)) * HD + h] = s;
      cnt += s;
    }
    __syncthreads();
  }
  atomicAdd(&counts[h], cnt);
}

// ---- Fold exact BatchNorm (binary input: var = mu*(1-mu)) into layer-2 input weights
__global__ void bnfold_kernel(const unsigned int* __restrict__ counts,
                              const float* __restrict__ gamma_, const float* __restrict__ beta_,
                              const float* __restrict__ w_ih2,
                              const float* __restrict__ b_ih2, const float* __restrict__ b_hh2,
                              float* __restrict__ w2eff, float* __restrict__ bias2)
{
  __shared__ float aS[HD], cS[HD];
  const int tid = threadIdx.x;
  if (tid < HD) {
    const float mu  = (float)counts[tid] / (float)(T_STEPS * LDIM);
    const float var = mu - mu * mu;                 // E[s^2]-E[s]^2 for s in {0,1}
    const float a   = gamma_[tid] * rsqrtf(var + 1e-5f);
    aS[tid] = a;
    cS[tid] = beta_[tid] - mu * a;
  }
  __syncthreads();
  for (int e = tid; e < G4 * HD; e += 256) w2eff[e] = w_ih2[e] * aS[e & (HD - 1)];
  for (int n = tid; n < G4; n += 256) {
    float s = b_ih2[n] + b_hh2[n];
    for (int k = 0; k < HD; ++k) s += w_ih2[(size_t)n * HD + k] * cS[k];
    bias2[n] = s;
  }
}

// ---- Layer 2: SLSTM scan on BN-folded spikes + time-mean of mem + final FC
__global__ __launch_bounds__(256) void slstm2_kernel(
    const unsigned char* __restrict__ spk1, // [T][L][128]
    const float* __restrict__ w2eff,        // [512][128]
    const float* __restrict__ w_hh2,        // [512][128]
    const float* __restrict__ bias2,        // [512]
    const float* __restrict__ thr2p,        // [1]
    const float* __restrict__ fc_w,         // [7][128]
    const float* __restrict__ fc_b,         // [7]
    float* __restrict__ out)                // [1024][7]
{
  __shared__ float At[TL * HD];
  __shared__ float MemS[TL * HD];
  __shared__ float SynS[TL * HD];
  __shared__ float Gates[TL * G4];

  const int tid = threadIdx.x;
  const int l0  = blockIdx.x * TL;
  const float thr = thr2p[0];

  for (int e = tid; e < TL * HD; e += 256) { MemS[e] = 0.f; SynS[e] = 0.f; }

  const int h  = tid & (HD - 1);
  const int rg = tid >> 7;
  const float bi = bias2[h];
  const float bf = bias2[h + HD];
  const float bg = bias2[h + 2 * HD];
  const float bo = bias2[h + 3 * HD];
  float msum[8];
#pragma unroll
  for (int jr = 0; jr < 8; ++jr) msum[jr] = 0.f;
  __syncthreads();

  for (int t = 0; t < T_STEPS; ++t) {
    // phase 1: unpack BN-ready binary spikes into At[TL][128]
    for (int j = 0; j < 8; ++j) {
      const int e  = tid + j * 256;       // 0..2047
      const int r  = e >> 7;
      const int hh = e & (HD - 1);
      At[e] = (float)spk1[((size_t)t * LDIM + (l0 + r)) * HD + hh];
    }
    __syncthreads();

    gemm_gates<HD>(At, MemS, w2eff, w_hh2, Gates);
    __syncthreads();

    for (int jr = 0; jr < 8; ++jr) {
      const int r = rg * 8 + jr;
      const float gi = Gates[r * G4 + h]          + bi;
      const float gf = Gates[r * G4 + h + HD]     + bf;
      const float gg = Gates[r * G4 + h + 2 * HD] + bg;
      const float go = Gates[r * G4 + h + 3 * HD] + bo;
      const float syn  = SynS[r * HD + h];
      const float memv = MemS[r * HD + h];
      const float sy = sigmoidf_(gf) * syn + sigmoidf_(gi) * tanhf(gg);
      const float mn = sigmoidf_(go) * tanhf(sy) - ((memv - thr > 0.f) ? thr : 0.f);
      SynS[r * HD + h] = sy;
      MemS[r * HD + h] = mn;
      msum[jr] += mn;
    }
    __syncthreads();
  }

  // time-mean of mem2 -> reuse At as final_mem[TL][128]
  for (int jr = 0; jr < 8; ++jr)
    At[(rg * 8 + jr) * HD + h] = msum[jr] * (1.0f / (float)T_STEPS);
  __syncthreads();

  // final FC: [TL][128] @ fc_w[7][128]^T + fc_b
  if (tid < TL * NCLS) {
    const int r  = tid / NCLS;
    const int nc = tid % NCLS;
    float acc = fc_b[nc];
    for (int hh = 0; hh < HD; ++hh) acc += At[r * HD + hh] * fc_w[nc * HD + hh];
    out[(size_t)(l0 + r) * NCLS + nc] = acc;
  }
}

extern "C" void kernel_launch(void* const* d_in, const int* in_sizes, int n_in,
                              void* d_out, int out_size, void* d_ws, size_t ws_size,
                              hipStream_t stream) {
  (void)in_sizes; (void)n_in; (void)out_size; (void)ws_size;
  const float* x      = (const float*)d_in[0];
  const float* conv_w = (const float*)d_in[1];
  const float* conv_b = (const float*)d_in[2];
  const float* w_ih1  = (const float*)d_in[3];
  const float* w_hh1  = (const float*)d_in[4];
  const float* b_ih1  = (const float*)d_in[5];
  const float* b_hh1  = (const float*)d_in[6];
  const float* thr1   = (const float*)d_in[7];
  const float* w_ih2  = (const float*)d_in[8];
  const float* w_hh2  = (const float*)d_in[9];
  const float* b_ih2  = (const float*)d_in[10];
  const float* b_hh2  = (const float*)d_in[11];
  const float* thr2   = (const float*)d_in[12];
  const float* bn_g   = (const float*)d_in[13];
  const float* bn_b   = (const float*)d_in[14];
  const float* fc_w   = (const float*)d_in[15];
  const float* fc_b   = (const float*)d_in[16];
  float* out = (float*)d_out;

  // workspace layout
  const size_t SPK_BYTES = (size_t)T_STEPS * LDIM * HD;        // 32 MB (u8 spikes)
  unsigned char* spk1   = (unsigned char*)d_ws;
  unsigned int*  counts = (unsigned int*)((char*)d_ws + SPK_BYTES);
  float*         w2eff  = (float*)((char*)d_ws + SPK_BYTES + 1024);
  float*         bias2  = (float*)((char*)d_ws + SPK_BYTES + 1024 + (size_t)G4 * HD * sizeof(float));

  const int nblocks = LDIM / TL;   // 64 persistent blocks

  zero_counts_kernel<<<1, 128, 0, stream>>>(counts);
  slstm1_kernel<<<nblocks, 256, 0, stream>>>(x, conv_w, conv_b, w_ih1, w_hh1,
                                             b_ih1, b_hh1, thr1, spk1, counts);
  bnfold_kernel<<<1, 256, 0, stream>>>(counts, bn_g, bn_b, w_ih2, b_ih2, b_hh2, w2eff, bias2);
  slstm2_kernel<<<nblocks, 256, 0, stream>>>(spk1, w2eff, w_hh2, bias2, thr2, fc_w, fc_b, out);
}